// leaky_integrator_VAE_worms_deterministic_68032281969206
// MI455X (gfx1250) — compile-verified
//
#include <hip/hip_runtime.h>
#include <hip/hip_bf16.h>

// ---------------------------------------------------------------------------
// CDNA5 (gfx1250) leaky-integrator RNN.
// t%400==0 resets -> 10 independent 400-step segments, batched as the N=16
// dim of V_WMMA_F32_16X16X4_F32 (exact fp32). Per-WG weight tiles (256 KB)
// are DMA'd into LDS once via the Tensor Data Mover and reused for all 400
// steps; state fragments are software-pipelined from L2. 128 persistent WGs,
// atomic grid barrier per step.
// ---------------------------------------------------------------------------

typedef float v2f  __attribute__((ext_vector_type(2)));
typedef float v4f  __attribute__((ext_vector_type(4)));
typedef float v8f  __attribute__((ext_vector_type(8)));
typedef unsigned int u32x4 __attribute__((ext_vector_type(4)));
typedef int   i32x4 __attribute__((ext_vector_type(4)));
typedef int   i32x8 __attribute__((ext_vector_type(8)));

#define NN    2048
#define TT    4000
#define SEG   400
#define NSEG  10          // active segments (batch), padded to 16
#define DTC   0.01f
#define NWG   128         // 2048 / 16 output-neuron tiles
#define NWAVE 8           // 256 threads
#define KB_PER_WAVE 64    // 2048 / (4 * NWAVE)
#define TILE_FLOATS (512 * 64)          // 32768 floats = 128 KB per matrix tile
#define LDS_DYN_BYTES (2 * TILE_FLOATS * 4)   // 256 KB

#ifndef USE_TDM
#if __has_builtin(__builtin_amdgcn_tensor_load_to_lds)
#define USE_TDM 1
#else
#define USE_TDM 0
#endif
#endif

// fragment index for state element X[k][s] (k = neuron, s = segment 0..15)
__device__ __forceinline__ int xfrag_idx(int k, int s) {
    int l = ((k & 2) << 3) + s;          // lane
    return (k >> 2) * 64 + l * 2 + (k & 1);
}

// ---------------------------------------------------------------------------
// Kernel A: build effective weights in WMMA A-fragment-major layout.
// Fragment float index i: r=i&1, lane=(i>>1)&31, kb=(i>>6)&511, tile mt=i>>15.
// A layout (16x4 fp32): lane l -> M=l&15, K=2*(l>>4)+r.
// ---------------------------------------------------------------------------
__global__ __launch_bounds__(256) void build_weights_kernel(
    const float* __restrict__ spC, const float* __restrict__ spE,
    const float* __restrict__ sgC, const float* __restrict__ magC,
    const float* __restrict__ magE,
    float* __restrict__ Wc, float* __restrict__ We)
{
    size_t i = (size_t)blockIdx.x * 256 + threadIdx.x;   // 0 .. 2048*2048-1
    int r  = (int)(i & 1);
    int l  = (int)((i >> 1) & 31);
    int kb = (int)((i >> 6) & 511);
    int mt = (int)(i >> 15);
    int m  = mt * 16 + (l & 15);
    int k  = kb * 4 + ((l >> 4) << 1) + r;
    size_t rm = (size_t)m * NN + k;      // row-major [m][k]
    size_t cm = (size_t)k * NN + m;      // transposed [k][m]
    Wc[i] = spC[rm] * fmaxf(magC[rm], 0.0f) * sgC[rm];
    We[i] = spE[rm] * (fmaxf(magE[rm], 0.0f) + fmaxf(magE[cm], 0.0f));
}

// ---------------------------------------------------------------------------
// Kernel B: W_e rowsums, initial batched state, boundary hs / rec(t=0).
// ---------------------------------------------------------------------------
__global__ __launch_bounds__(256) void init_state_kernel(
    const float* __restrict__ spE, const float* __restrict__ magE,
    const float* __restrict__ infer,
    float* __restrict__ rowsum, float* __restrict__ X0, float* __restrict__ X1,
    float* __restrict__ out)
{
    int m = blockIdx.x;
    int tid = threadIdx.x;
    float acc = 0.0f;
    for (int k = tid; k < NN; k += 256) {
        size_t rm = (size_t)m * NN + k;
        size_t cm = (size_t)k * NN + m;
        acc += spE[rm] * (fmaxf(magE[rm], 0.0f) + fmaxf(magE[cm], 0.0f));
    }
    __shared__ float red[256];
    red[tid] = acc;
    __syncthreads();
    for (int off = 128; off > 0; off >>= 1) {
        if (tid < off) red[tid] += red[tid + off];
        __syncthreads();
    }
    if (tid == 0) rowsum[m] = red[0];

    if (tid < 16) {
        int s = tid;
        int xi = xfrag_idx(m, s);
        float v = (s < NSEG) ? infer[(size_t)m * TT + s * SEG] : 0.0f;
        X0[xi] = v;           // state after p = 0  (reset to infer)
        X1[xi] = 0.0f;
        if (s < NSEG) out[(size_t)m * TT + s * SEG] = v;               // hs at resets
        if (s == 0)  out[(size_t)NN * TT + (size_t)m * TT] = 0.0f;     // rec(t=0)=0
    }
}

// ---------------------------------------------------------------------------
// Grid-wide generation barrier (state in d_ws, memset to 0 each launch).
// ---------------------------------------------------------------------------
__device__ __forceinline__ void grid_barrier(unsigned* cnt, unsigned* gen,
                                             unsigned nwg)
{
    __syncthreads();
    if (threadIdx.x == 0) {
        __threadfence();
        unsigned g = __hip_atomic_load(gen, __ATOMIC_RELAXED,
                                       __HIP_MEMORY_SCOPE_AGENT);
        unsigned a = __hip_atomic_fetch_add(cnt, 1u, __ATOMIC_ACQ_REL,
                                            __HIP_MEMORY_SCOPE_AGENT);
        if (a == nwg - 1u) {
            __hip_atomic_store(cnt, 0u, __ATOMIC_RELAXED,
                               __HIP_MEMORY_SCOPE_AGENT);
            __hip_atomic_fetch_add(gen, 1u, __ATOMIC_ACQ_REL,
                                   __HIP_MEMORY_SCOPE_AGENT);
        } else {
            while (__hip_atomic_load(gen, __ATOMIC_ACQUIRE,
                                     __HIP_MEMORY_SCOPE_AGENT) == g) {
                __builtin_amdgcn_s_sleep(1);
            }
        }
        __threadfence();
    }
    __syncthreads();
}

#if USE_TDM
// ---------------------------------------------------------------------------
// One-shot Tensor Data Mover copy: 128 KB (16384 x 8B) global -> LDS.
// D# group layout per CDNA5 ISA ch.8 (1D tile, data_size=8B, no multicast).
// ---------------------------------------------------------------------------
__device__ __forceinline__ void tdm_load_tile(const float* gsrc, float* ldst)
{
    unsigned long long ga = (unsigned long long)(uintptr_t)gsrc;
    unsigned lo = (unsigned)(uintptr_t)ldst;       // LDS byte offset = addr[31:0]
    u32x4 g0;
    g0.x = 1u;                                     // count=1 (valid descriptor)
    g0.y = lo;                                     // lds_addr
    g0.z = (unsigned)(ga & 0xffffffffull);         // global_addr[31:0]
    g0.w = (unsigned)((ga >> 32) & 0x01ffffffull)  // global_addr[56:32]
         | 0x80000000u;                            // type = 2 ("image")
    i32x8 g1 = {};
    g1[0] = 0x00030000;     // workgroup_mask=0, data_size=3 (8-byte units)
    g1[1] = 0x40000000;     // tensor_dim0[15:0] = 16384  (bits[63:48])
    g1[2] = 0x00010000;     // tensor_dim0 hi = 0, tensor_dim1 = 1
    g1[3] = 0x40000000;     // tile_dim0 = 16384          (bits[127:112])
    g1[4] = 0;              // tile_dim1 = tile_dim2 = 0 (unused)
    g1[5] = 0x4000;         // tensor_dim0_stride = 16384
    i32x4 g2 = {};
    i32x4 g3 = {};
#if __clang_major__ >= 23
    i32x8 g4 = {};
    __builtin_amdgcn_tensor_load_to_lds(g0, g1, g2, g3, g4, 0);
#else
    __builtin_amdgcn_tensor_load_to_lds(g0, g1, g2, g3, 0);
#endif
}
#endif

// ---------------------------------------------------------------------------
// Persistent recurrence kernel. blockIdx.x = output-neuron tile (16 rows).
// p = 1..399 : full leaky step via 2 x 512 fp32 WMMA (weights from LDS).
// p = 400    : boundary rec patch from the final state of each segment.
// ---------------------------------------------------------------------------
__global__ __launch_bounds__(256, 1) void worm_loop_kernel(
    const float* __restrict__ in0,     // (1,N,T) external input
    const float* __restrict__ bias,
    const float* __restrict__ tau,
    const float* __restrict__ Wc,      // fragment-major
    const float* __restrict__ We,      // fragment-major
    const float* __restrict__ rowsum,
    float* __restrict__ X0, float* __restrict__ X1,   // ping-pong state (frag)
    float* __restrict__ out,
    unsigned* __restrict__ bar_cnt, unsigned* __restrict__ bar_gen)
{
    const int wg   = blockIdx.x;
    const int tid  = threadIdx.x;
    const int lane = tid & 31;
    const int wave = tid >> 5;

    extern __shared__ float lds[];                 // 256 KB dynamic
    float* ldsWc = lds;
    float* ldsWe = lds + TILE_FLOATS;
    __shared__ float sC[NWAVE][256];
    __shared__ float sE[NWAVE][256];

    // ---- one-time weight-tile preload into LDS (reused for all 400 steps) --
    const float* gWc = Wc + (size_t)wg * TILE_FLOATS;
    const float* gWe = We + (size_t)wg * TILE_FLOATS;
#if USE_TDM
    if (tid < 32) {                                // wave 0 issues the DMAs
        tdm_load_tile(gWc, ldsWc);
        tdm_load_tile(gWe, ldsWe);
        __builtin_amdgcn_s_wait_tensorcnt(0);
    }
#else
    for (int i = tid; i < TILE_FLOATS / 4; i += 256) {
        ((v4f*)ldsWc)[i] = ((const v4f*)gWc)[i];
        ((v4f*)ldsWe)[i] = ((const v4f*)gWe)[i];
    }
#endif
    __syncthreads();

    // per-thread epilogue element: (m_loc, s)
    const int m_loc = tid >> 4;
    const int s     = tid & 15;
    const int m     = wg * 16 + m_loc;
    const float tau_m  = tau[m];
    const float bias_m = bias[m];
    const float rs_m   = rowsum[m];
    const int   xi     = xfrag_idx(m, s);

    const float* lwc = ldsWc + wave * (KB_PER_WAVE * 64) + lane * 2;
    const float* lwe = ldsWe + wave * (KB_PER_WAVE * 64) + lane * 2;

    for (int p = 1; p <= SEG; ++p) {
        const float* __restrict__ Xprev = (p & 1) ? X0 : X1;  // state after p-1
        float*       __restrict__ Xnext = (p & 1) ? X1 : X0;  // state after p

        // hoisted epilogue operands (latency hidden under the matmul)
        float prevx = Xprev[xi];
        const int t = s * SEG + p;
        float u = 0.0f;
        if (p < SEG && s < NSEG) u = in0[(size_t)m * TT + t];

        const float* xbase = Xprev + (size_t)wave * (KB_PER_WAVE * 64) + lane * 2;

        v8f accC = {};
        v8f accE = {};
        v2f xb[2][8];                      // double-buffered state fragments
#pragma unroll
        for (int j = 0; j < 8; ++j)
            xb[0][j] = *(const v2f*)(xbase + j * 64);

#pragma unroll
        for (int blk = 0; blk < 8; ++blk) {
            const int cur = blk & 1;
            if (blk < 7) {                 // prefetch next 8 fragments
#pragma unroll
                for (int j = 0; j < 8; ++j)
                    xb[cur ^ 1][j] =
                        *(const v2f*)(xbase + ((blk + 1) * 8 + j) * 64);
            }
#pragma unroll
            for (int j = 0; j < 8; ++j) {
                const int kb = blk * 8 + j;
                v2f a_c = *(const v2f*)(lwc + kb * 64);   // ds_load_b64
                v2f a_e = *(const v2f*)(lwe + kb * 64);
                v2f b   = xb[cur][j];
                v2f br  = { fmaxf(b.x, 0.0f), fmaxf(b.y, 0.0f) };
                accC = __builtin_amdgcn_wmma_f32_16x16x4_f32(
                           false, a_c, false, br, (short)0, accC, false, false);
                accE = __builtin_amdgcn_wmma_f32_16x16x4_f32(
                           false, a_e, false, b,  (short)0, accE, false, false);
            }
        }

        // scatter C/D fragment (vgpr r, lane l) -> element (r + 8*(l>>4), l&15)
#pragma unroll
        for (int r = 0; r < 8; ++r) {
            int e = (r + ((lane >> 4) << 3)) * 16 + (lane & 15);
            sC[wave][e] = accC[r];
            sE[wave][e] = accE[r];
        }
        __syncthreads();

        float chem = 0.0f, elecw = 0.0f;
#pragma unroll
        for (int w = 0; w < NWAVE; ++w) {
            chem  += sC[w][tid];
            elecw += sE[w][tid];
        }
        float rec = chem + elecw - prevx * rs_m;
        __syncthreads();   // LDS reuse next iteration

        if (p < SEG) {
            float newx = 0.0f;
            if (s < NSEG) {
                newx = (DTC / tau_m) * (rec + bias_m + u - prevx) + prevx;
                out[(size_t)m * TT + t]                   = newx;  // hs
                out[(size_t)NN * TT + (size_t)m * TT + t] = rec;   // recs
            }
            Xnext[xi] = newx;
            grid_barrier(bar_cnt, bar_gen, gridDim.x);
        } else {
            // boundary patch: rec at t=(s+1)*400 uses final state of segment s
            if (s < NSEG - 1) {
                out[(size_t)NN * TT + (size_t)m * TT + (s + 1) * SEG] = rec;
            }
            // last iteration: no state write, no barrier needed
        }
    }
}

// ---------------------------------------------------------------------------
// Host launcher (graph-capture safe: only async stream ops).
// ---------------------------------------------------------------------------
extern "C" void kernel_launch(void* const* d_in, const int* in_sizes, int n_in,
                              void* d_out, int out_size, void* d_ws, size_t ws_size,
                              hipStream_t stream)
{
    const float* in0   = (const float*)d_in[0];   // input (1,N,T)
    const float* infer = (const float*)d_in[1];   // hidden_infer_states (1,N,T)
    const float* spC   = (const float*)d_in[2];
    const float* spE   = (const float*)d_in[3];
    const float* sgC   = (const float*)d_in[4];
    const float* magC  = (const float*)d_in[5];
    const float* magE  = (const float*)d_in[6];
    const float* bias  = (const float*)d_in[7];
    const float* tau   = (const float*)d_in[8];
    float* out = (float*)d_out;

    // d_ws layout (floats): Wc[4M] | We[4M] | rowsum[2048] | X0[32768] |
    //                       X1[32768] | barrier[2 u32]      (~34 MB total)
    float* ws      = (float*)d_ws;
    float* Wc      = ws;
    float* We      = Wc + (size_t)NN * NN;
    float* rowsum  = We + (size_t)NN * NN;
    float* X0      = rowsum + NN;
    float* X1      = X0 + NN * 16;
    unsigned* bar  = (unsigned*)(X1 + NN * 16);

    (void)hipFuncSetAttribute((const void*)worm_loop_kernel,
                              hipFuncAttributeMaxDynamicSharedMemorySize,
                              LDS_DYN_BYTES);

    hipMemsetAsync(bar, 0, 2 * sizeof(unsigned), stream);

    build_weights_kernel<<<(NN * NN) / 256, 256, 0, stream>>>(
        spC, spE, sgC, magC, magE, Wc, We);

    init_state_kernel<<<NN, 256, 0, stream>>>(
        spE, magE, infer, rowsum, X0, X1, out);

    worm_loop_kernel<<<NWG, 256, LDS_DYN_BYTES, stream>>>(
        in0, bias, tau, Wc, We, rowsum, X0, X1, out, bar, bar + 1);
}